// JKNet_68719476736446
// MI455X (gfx1250) — compile-verified
//
#include <hip/hip_runtime.h>
#include <hip/hip_bf16.h>

typedef __attribute__((ext_vector_type(2))) float v2f;
typedef __attribute__((ext_vector_type(8))) float v8f;

#define NN 100000
#define NE 3200000
#define KIN 1433
#define HID 16
#define NCLS 7
#define NLAY 6

// ---------------------------------------------------------------------------
// fill a float buffer with a constant
__global__ void fill_kernel(float* __restrict__ p, float v, long long n) {
    long long g = (long long)blockIdx.x * blockDim.x + threadIdx.x;
    if (g < n) p[g] = v;
}

// degree accumulation: deg[dst] += 1 for each edge (deg pre-filled with 1.0
// to account for the self-loop)
__global__ void deg_kernel(const int* __restrict__ ei, float* __restrict__ deg, int E) {
    int e = blockIdx.x * blockDim.x + threadIdx.x;
    if (e < E) atomicAdd(&deg[ei[E + e]], 1.0f);
}

// dinv = rsqrt(max(deg, 1)) in place
__global__ void dinv_kernel(float* __restrict__ d, int n) {
    int i = blockIdx.x * blockDim.x + threadIdx.x;
    if (i < n) d[i] = rsqrtf(fmaxf(d[i], 1.0f));
}

// ---------------------------------------------------------------------------
// C[M x 16] = A[M x K] * B[K x 16], one wave32 per 16-row tile, using
// V_WMMA_F32_16X16X4_F32. K arbitrary (1433 for layer 0, 16 for hidden).
__global__ void wmma_gemm_kernel(const float* __restrict__ A,
                                 const float* __restrict__ B,
                                 float* __restrict__ C,
                                 int nrows, int K) {
    const int lane = threadIdx.x & 31;
    const int tile = blockIdx.x * (blockDim.x >> 5) + (threadIdx.x >> 5);
    if (tile * 16 >= nrows) return;          // wave-uniform: EXEC stays full
    const int row0 = tile * 16;
    const int m  = lane & 15;                // A: row, B: column
    const int kq = (lane >> 4) * 2;          // K sub-pair selected by half-wave

    const float* arow = A + (size_t)(row0 + m) * K;
    v8f c = {};

    int k = 0;
    for (; k + 4 <= K; k += 4) {
        v2f a, b;
        a.x = arow[k + kq];
        a.y = arow[k + kq + 1];
        b.x = B[(size_t)(k + kq)     * 16 + m];
        b.y = B[(size_t)(k + kq + 1) * 16 + m];
        c = __builtin_amdgcn_wmma_f32_16x16x4_f32(false, a, false, b,
                                                  (short)0, c, false, false);
    }
    if (k < K) {                             // K tail (K=1433 -> 1 column)
        const int k0 = k + kq, k1 = k + kq + 1;
        v2f a, b;
        a.x = (k0 < K) ? arow[k0] : 0.0f;
        a.y = (k1 < K) ? arow[k1] : 0.0f;
        b.x = (k0 < K) ? B[(size_t)k0 * 16 + m] : 0.0f;
        b.y = (k1 < K) ? B[(size_t)k1 * 16 + m] : 0.0f;
        c = __builtin_amdgcn_wmma_f32_16x16x4_f32(false, a, false, b,
                                                  (short)0, c, false, false);
    }

    // D layout: VGPR r, lanes 0-15 -> row r, lanes 16-31 -> row r+8; col = m
    const int rbase = (lane >> 4) * 8;
#pragma unroll
    for (int r = 0; r < 8; ++r)
        C[(size_t)(row0 + rbase + r) * 16 + m] = c[r];
}

// ---------------------------------------------------------------------------
// out[dst] += h[src] * dinv[src] * dinv[dst], 16 lanes per edge
// (coalesced 64B gather of h row, cacheline-coalesced f32 atomics on out row)
__global__ void agg_kernel(const float* __restrict__ h,
                           const int* __restrict__ ei,
                           const float* __restrict__ dinv,
                           float* __restrict__ out,
                           int E, int N) {
    long long g = (long long)blockIdx.x * blockDim.x + threadIdx.x;
    long long total = (long long)(E + N) * 16;
    if (g >= total) return;
    const int j = (int)(g & 15);
    const long long e = g >> 4;
    int s, d;
    if (e < (long long)E) { s = ei[e]; d = ei[E + e]; }
    else                  { s = d = (int)(e - E); }   // self loop
    const float nrm = dinv[s] * dinv[d];
    atomicAdd(&out[(size_t)d * 16 + j], h[(size_t)s * 16 + j] * nrm);
}

// h = (agg + b) [relu]; also scatter into JK concat buffer at column col0
__global__ void bias_act_kernel(const float* __restrict__ agg,
                                const float* __restrict__ b,
                                float* __restrict__ h,
                                float* __restrict__ hcat,
                                int N, int col0, int doRelu) {
    int g = blockIdx.x * blockDim.x + threadIdx.x;
    if (g >= N * 16) return;
    const int n = g >> 4, j = g & 15;
    float v = agg[g] + b[j];
    if (doRelu) v = fmaxf(v, 0.0f);
    h[g] = v;
    hcat[(size_t)n * (NLAY * HID) + col0 + j] = v;
}

// logits = hcat @ Wlin + blin; out = log_softmax(logits)
__global__ void final_kernel(const float* __restrict__ hcat,
                             const float* __restrict__ Wlin,
                             const float* __restrict__ blin,
                             float* __restrict__ out, int N) {
    int n = blockIdx.x * blockDim.x + threadIdx.x;
    if (n >= N) return;
    const float* hr = hcat + (size_t)n * (NLAY * HID);
    float logit[NCLS];
#pragma unroll
    for (int c = 0; c < NCLS; ++c) logit[c] = blin[c];
    for (int k = 0; k < NLAY * HID; ++k) {
        const float hv = hr[k];
#pragma unroll
        for (int c = 0; c < NCLS; ++c) logit[c] += hv * Wlin[k * NCLS + c];
    }
    float mx = logit[0];
#pragma unroll
    for (int c = 1; c < NCLS; ++c) mx = fmaxf(mx, logit[c]);
    float se = 0.0f;
#pragma unroll
    for (int c = 0; c < NCLS; ++c) se += __expf(logit[c] - mx);
    const float lse = mx + __logf(se);
#pragma unroll
    for (int c = 0; c < NCLS; ++c) out[(size_t)n * NCLS + c] = logit[c] - lse;
}

// ---------------------------------------------------------------------------
extern "C" void kernel_launch(void* const* d_in, const int* in_sizes, int n_in,
                              void* d_out, int out_size, void* d_ws, size_t ws_size,
                              hipStream_t stream) {
    const float* x    = (const float*)d_in[0];
    const int*   ei   = (const int*)  d_in[1];
    const float* W0   = (const float*)d_in[2];
    const float* b0   = (const float*)d_in[3];
    const float* Ws   = (const float*)d_in[4];   // [NLAY-1, 16, 16]
    const float* bs   = (const float*)d_in[5];   // [NLAY-1, 16]
    const float* Wlin = (const float*)d_in[6];   // [96, 7]
    const float* blin = (const float*)d_in[7];   // [7]
    float* out = (float*)d_out;

    // workspace carve (256B aligned)
    char* wsp = (char*)d_ws;
    auto carve = [&](size_t bytes) {
        float* p = (float*)wsp;
        wsp += (bytes + 255) & ~(size_t)255;
        return p;
    };
    float* dinv = carve((size_t)NN * 4);
    float* tmp  = carve((size_t)NN * HID * 4);
    float* aggb = carve((size_t)NN * HID * 4);
    float* h    = carve((size_t)NN * HID * 4);
    float* hcat = carve((size_t)NN * HID * NLAY * 4);

    const int T = 256;

    // --- symmetric GCN normalization ---
    fill_kernel<<<(NN + T - 1) / T, T, 0, stream>>>(dinv, 1.0f, (long long)NN);
    deg_kernel<<<(NE + T - 1) / T, T, 0, stream>>>(ei, dinv, NE);
    dinv_kernel<<<(NN + T - 1) / T, T, 0, stream>>>(dinv, NN);

    const int tiles = NN / 16;                       // 6250 wave-tiles
    const int wavesPerBlock = T / 32;                // 8
    const int gemmBlocks = (tiles + wavesPerBlock - 1) / wavesPerBlock;
    const long long aggThreads = (long long)(NE + NN) * 16;
    const int aggBlocks = (int)((aggThreads + T - 1) / T);
    const int nhBlocks = (NN * HID + T - 1) / T;

    // --- layer 0: h = agg(x @ W0) + b0 ---
    wmma_gemm_kernel<<<gemmBlocks, T, 0, stream>>>(x, W0, tmp, NN, KIN);
    fill_kernel<<<nhBlocks, T, 0, stream>>>(aggb, 0.0f, (long long)NN * HID);
    agg_kernel<<<aggBlocks, T, 0, stream>>>(tmp, ei, dinv, aggb, NE, NN);
    bias_act_kernel<<<nhBlocks, T, 0, stream>>>(aggb, b0, h, hcat, NN, 0, 0);

    // --- hidden layers: h = relu(agg(h @ Ws[i]) + bs[i]) ---
    for (int l = 1; l < NLAY; ++l) {
        wmma_gemm_kernel<<<gemmBlocks, T, 0, stream>>>(h, Ws + (size_t)(l - 1) * HID * HID,
                                                       tmp, NN, HID);
        fill_kernel<<<nhBlocks, T, 0, stream>>>(aggb, 0.0f, (long long)NN * HID);
        agg_kernel<<<aggBlocks, T, 0, stream>>>(tmp, ei, dinv, aggb, NE, NN);
        bias_act_kernel<<<nhBlocks, T, 0, stream>>>(aggb, bs + (size_t)(l - 1) * HID,
                                                    h, hcat, NN, l * HID, 1);
    }

    // --- JK concat -> linear -> log_softmax ---
    final_kernel<<<(NN + T - 1) / T, T, 0, stream>>>(hcat, Wlin, blin, out, NN);
}